// Global_APro_11871289606747
// MI455X (gfx1250) — compile-verified
//
#include <hip/hip_runtime.h>

#define BS   8
#define CCH  256
#define HH   96
#define WW   96
#define NN   (HH * WW)   /* 9216 */
#define EE   (NN - 1)    /* 9215 */
#define ZETA 0.01f

typedef __attribute__((ext_vector_type(2))) float v2f;
typedef __attribute__((ext_vector_type(8))) float v8f;

// ---------------------------------------------------------------------------
// Kernel A: edge weights via V_WMMA_F32_16X16X4_F32 (fp32 matrix pipe).
// One wave handles 16 edges: build diff rows D[16][256] on the fly and
// accumulate the Gram matrix C += A * A^T in 64 steps of K=4.  diag(C) are
// the squared distances.  A(16x4) and B(4x16)=A^T share the same VGPR layout,
// so one v2f per lane feeds both operands.
// ---------------------------------------------------------------------------
__global__ __launch_bounds__(256) void edge_weight_wmma(
    const float* __restrict__ emb,   // [BS, C, N]
    const int*   __restrict__ tree,  // [BS, E, 2] (src=parent, tgt=child)
    float*       __restrict__ wgt)   // [BS, E]
{
  const int GROUPS = (EE + 15) / 16;                 // 576 groups / batch
  int gg   = blockIdx.x * 8 + (threadIdx.x >> 5);    // global group id
  int b    = gg / GROUPS;
  int g    = gg - b * GROUPS;
  int lane = threadIdx.x & 31;
  int m    = lane & 15;           // edge-in-group (row of A, col of B)
  int kb   = (lane >> 4) * 2;     // K offset inside each 4-chunk

  int e  = g * 16 + m;
  int ec = e < EE ? e : EE - 1;   // clamp pad lanes (no divergence)
  const int* tb = tree + (size_t)b * EE * 2 + 2 * ec;
  int s = tb[0];
  int t = tb[1];

  const float* p = emb + ((size_t)b * CCH + kb) * NN;
  const float* pS = p + s;
  const float* pT = p + t;

  v8f c = {};
#pragma unroll 4
  for (int k0 = 0; k0 < CCH; k0 += 4) {
    v2f a;
    a.x = pS[0]  - pT[0];     // channel k0+kb
    a.y = pS[NN] - pT[NN];    // channel k0+kb+1
    // D = A*B + C ; B register == A register (Gram matrix)
    c = __builtin_amdgcn_wmma_f32_16x16x4_f32(false, a, false, a,
                                              (short)0, c, false, false);
    pS += (size_t)4 * NN;
    pT += (size_t)4 * NN;
  }

  // Diagonal extraction: (m,m) at lane m comp m (m<8), lane m+16 comp m-8.
  bool own = false; int md = 0, comp = 0;
  if (lane < 8)        { own = true; md = lane;      comp = lane;      }
  else if (lane >= 24) { own = true; md = lane - 16; comp = lane - 24; }
  float dist = 0.0f;
#pragma unroll
  for (int j = 0; j < 8; ++j)
    if (j == comp) dist = c[j];

  if (own) {
    int eo = g * 16 + md;
    if (eo < EE) wgt[(size_t)b * EE + eo] = __expf(-ZETA * dist);
  }
}

// ---------------------------------------------------------------------------
// Kernel B: level-scheduled tree DP.  One 1024-thread workgroup per batch.
// Depth via pointer doubling (14 iters), counting-sort nodes by depth, then
// up pass (ds_add_f32 atomics into parents) and down pass per level with
// workgroup barriers.  Two runs: F (init=f) kept in registers, G (init=1).
// LDS is time-multiplexed to stay under 64KB.
// ---------------------------------------------------------------------------
#define NT   1024
#define NPT  (NN / NT)   /* 9 nodes per thread, exact */

__global__ __launch_bounds__(NT) void tree_scan(
    const float* __restrict__ feat,  // [BS, N]
    const int*   __restrict__ tree,  // [BS, E, 2]
    const float* __restrict__ wgt,   // [BS, E]
    float*       __restrict__ out)   // [BS, N]
{
  __shared__ __align__(16) char buf[57344];
  __shared__ int maxd_sh;

  int b   = blockIdx.x;
  int tid = threadIdx.x;
  const int*   tb = tree + (size_t)b * EE * 2;
  const float* wb = wgt  + (size_t)b * EE;
  const float* fb = feat + (size_t)b * NN;

  int*            anc   = (int*)buf;                       //  0    .. 36863
  unsigned char*  dep   = (unsigned char*)(buf + 36864);   // 36864 .. 46079
  unsigned short* order = (unsigned short*)buf;            //  0    .. 18431 (anc dead)
  int*            lend  = (int*)(buf + 18432);             // level end offsets
  int*            lptr  = (int*)(buf + 19456);             // scatter cursors
  float*          S     = (float*)(buf + 20480);           // 20480 .. 57343 (dep dead)

  // init parent links + unit depths
  for (int i = 0; i < NPT; ++i) {
    int v = tid + i * NT;
    anc[v] = (v == 0) ? 0 : tb[2 * (v - 1)];
    dep[v] = (v == 0) ? 0 : 1;
  }
  if (tid == 0) maxd_sh = 0;
  __syncthreads();

  // pointer doubling: 2^14 > N, read-all / barrier / write-all per iter
  for (int it = 0; it < 14; ++it) {
    int aa[NPT], dd[NPT];
    for (int i = 0; i < NPT; ++i) {
      int v = tid + i * NT;
      int a = anc[v];
      dd[i] = dep[a];
      aa[i] = anc[a];
    }
    __syncthreads();
    for (int i = 0; i < NPT; ++i) {
      int v  = tid + i * NT;
      int nd = (int)dep[v] + dd[i];
      dep[v] = (unsigned char)(nd > 255 ? 255 : nd);
      anc[v] = aa[i];
    }
    __syncthreads();
  }

  // histogram by depth
  for (int l = tid; l < 256; l += NT) lend[l] = 0;
  __syncthreads();
  int mydep[NPT];
  int mymax = 0;
  for (int i = 0; i < NPT; ++i) {
    mydep[i] = dep[tid + i * NT];
    atomicAdd(&lend[mydep[i]], 1);
    mymax = mydep[i] > mymax ? mydep[i] : mymax;
  }
  atomicMax(&maxd_sh, mymax);
  __syncthreads();
  if (tid == 0) {                       // inclusive scan -> level end offsets
    int run = 0;
    for (int l = 0; l < 256; ++l) { run += lend[l]; lend[l] = run; }
  }
  __syncthreads();
  for (int l = tid; l < 256; l += NT) lptr[l] = l ? lend[l - 1] : 0;
  __syncthreads();
  for (int i = 0; i < NPT; ++i) {       // counting-sort scatter
    int v   = tid + i * NT;
    int pos = atomicAdd(&lptr[mydep[i]], 1);
    order[pos] = (unsigned short)v;
  }
  __syncthreads();
  int MD = maxd_sh;

  float freg[NPT];
  for (int run = 0; run < 2; ++run) {
    for (int i = 0; i < NPT; ++i) {
      int v = tid + i * NT;
      S[v] = (run == 0) ? fb[v] : 1.0f;
    }
    __syncthreads();

    // up pass: leaves -> root, ds_add_f32 into parents (level L-1)
    for (int L = MD; L >= 1; --L) {
      int st = lend[L - 1], en = lend[L];
      for (int idx = st + tid; idx < en; idx += NT) {
        int v = order[idx];
        int ei = v - 1;
        int pq = tb[2 * ei];
        float w = wb[ei];
        atomicAdd(&S[pq], w * S[v]);
      }
      __syncthreads();
    }

    // down pass: root -> leaves, unique writes per node
    for (int L = 1; L <= MD; ++L) {
      int st = lend[L - 1], en = lend[L];
      for (int idx = st + tid; idx < en; idx += NT) {
        int v = order[idx];
        int ei = v - 1;
        int pq = tb[2 * ei];
        float w = wb[ei];
        S[v] = w * S[pq] + (1.0f - w * w) * S[v];
      }
      __syncthreads();
    }

    if (run == 0) {
      for (int i = 0; i < NPT; ++i) freg[i] = S[tid + i * NT];
      __syncthreads();             // protect S before G-run re-init
    } else {
      for (int i = 0; i < NPT; ++i) {
        int v = tid + i * NT;
        out[(size_t)b * NN + v] = freg[i] / S[v];
      }
    }
  }
}

// ---------------------------------------------------------------------------
extern "C" void kernel_launch(void* const* d_in, const int* in_sizes, int n_in,
                              void* d_out, int out_size, void* d_ws, size_t ws_size,
                              hipStream_t stream) {
  const float* feat = (const float*)d_in[0];   // feature_in  [8,1,96,96] f32
  const float* emb  = (const float*)d_in[1];   // embed_in    [8,256,96,96] f32
  const int*   tree = (const int*)d_in[2];     // tree        [8,9215,2] i32
  float* out = (float*)d_out;                  // [8,1,96,96] f32
  float* wgt = (float*)d_ws;                   // scratch: 8*9215 floats (~288KB)

  const int GROUPS = (EE + 15) / 16;           // 576
  int blocksA = (BS * GROUPS) / 8;             // 576 blocks of 8 waves
  edge_weight_wmma<<<blocksA, 256, 0, stream>>>(emb, tree, wgt);
  tree_scan<<<BS, NT, 0, stream>>>(feat, tree, wgt, out);
}